// TokenMixWeightGenerator_84911503442796
// MI455X (gfx1250) — compile-verified
//
#include <hip/hip_runtime.h>

#define LSEQ 2048
#define DMODEL 512
#define NHEAD 8
#define DH 64
#define NBH 16          // BATCH * NHEAD
#define EPSV 1e-8f

typedef float v2f __attribute__((ext_vector_type(2)));
typedef float v8f __attribute__((ext_vector_type(8)));

// ---------------------------------------------------------------------------
// Pass 1: sequential cumprod scan per (b,h,d) channel.
//   num[l] = cumA[l-1] (num[0]=1), den[l] = cumA[l-1]+eps (den[0]=1)
//   Anum[bh,l,d] = C * num ; Bdiv[bh,l,d] = B / den   (layout: d fastest)
// 64 threads per block = one head-dim slice; loads/stores coalesce across d.
// ---------------------------------------------------------------------------
__global__ __launch_bounds__(64) void tokenmix_prepass(
    const float* __restrict__ Bi, const float* __restrict__ Ci,
    const float* __restrict__ Si, float* __restrict__ Anum,
    float* __restrict__ Bdiv) {
  const int d  = threadIdx.x;           // 0..63
  const int bh = blockIdx.x;            // 0..15
  const int b  = bh >> 3;
  const int h  = bh & 7;
  size_t in_idx = (size_t)b * LSEQ * DMODEL + (size_t)h * DH + d;
  size_t ws_idx = ((size_t)bh * LSEQ) * DH + d;
  float prev = 1.0f;                    // cumA[l-1]
  for (int l = 0; l < LSEQ; ++l) {
    float s = Si[in_idx];
    float c = Ci[in_idx];
    float x = Bi[in_idx];
    float num = prev;
    float den = (l == 0) ? 1.0f : (prev + EPSV);
    Anum[ws_idx] = c * num;
    Bdiv[ws_idx] = x / den;
    prev *= s;
    in_idx += DMODEL;
    ws_idx += DH;
  }
}

// ---------------------------------------------------------------------------
// Pass 2: causal GEMM  T[bh] = Anum[bh] @ Bdiv[bh]^T  via v_wmma_f32_16x16x4_f32
// Block: 256 threads (8 waves), 128(i) x 64(j) output tile.
// Wave: 32x32 region = 2x2 accumulators with A/B fragment reuse.
// LDS staging with 66-float row pitch to avoid bank conflicts.
// ---------------------------------------------------------------------------
__global__ __launch_bounds__(256) void tokenmix_gemm(
    const float* __restrict__ Bi, const float* __restrict__ Ci,
    const float* __restrict__ Anum, const float* __restrict__ Bdiv,
    float* __restrict__ Out) {
  __shared__ float As[128 * 66];
  __shared__ float Bs[64 * 66];

  const int bh = blockIdx.z;
  const int ib = blockIdx.y * 128;      // block row base
  const int jb = blockIdx.x * 64;       // block col base
  float* __restrict__ outp = Out + (size_t)bh * LSEQ * LSEQ;

  // Block entirely above the diagonal -> pure zero fill, no GEMM, no staging.
  if (jb > ib + 127) {
    const float4 z = make_float4(0.f, 0.f, 0.f, 0.f);
    for (int t = threadIdx.x; t < 128 * 16; t += 256) {
      const int r = t >> 4, c4 = t & 15;
      *(float4*)&outp[(size_t)(ib + r) * LSEQ + jb + c4 * 4] = z;
    }
    return;
  }

  // Stage contiguous 128x64 A-tile and 64x64 B-tile into padded LDS.
  const float4* __restrict__ Ag =
      (const float4*)(Anum + ((size_t)bh * LSEQ + ib) * DH);
  const float4* __restrict__ Bg =
      (const float4*)(Bdiv + ((size_t)bh * LSEQ + jb) * DH);
  for (int t = threadIdx.x; t < 128 * 16; t += 256) {
    const int r = t >> 4, c4 = t & 15;
    *(float4*)&As[r * 66 + c4 * 4] = Ag[t];
  }
  for (int t = threadIdx.x; t < 64 * 16; t += 256) {
    const int r = t >> 4, c4 = t & 15;
    *(float4*)&Bs[r * 66 + c4 * 4] = Bg[t];
  }
  __syncthreads();

  const int lane = threadIdx.x & 31;
  const int w    = threadIdx.x >> 5;
  const int half = lane >> 4;           // 0: lanes 0-15, 1: lanes 16-31
  const int m    = lane & 15;
  const int ia   = (w & 3) * 32;        // wave row offset inside block tile
  const int ja   = (w >> 2) * 32;       // wave col offset inside block tile
  const int i0   = ib + ia;             // global (multiple of 32)
  const int j0   = jb + ja;             // global (multiple of 32)

  v8f acc00 = {}, acc01 = {}, acc10 = {}, acc11 = {};

  if (j0 <= i0 + 31) {                  // wave-uniform: some lower part exists
    const float* __restrict__ Ar0 = &As[(ia + m) * 66];
    const float* __restrict__ Ar1 = &As[(ia + 16 + m) * 66];
    const float* __restrict__ Br0 = &Bs[(ja + m) * 66];
    const float* __restrict__ Br1 = &Bs[(ja + 16 + m) * 66];
#pragma unroll
    for (int kk = 0; kk < DH; kk += 4) {
      const int ko = kk + half * 2;     // A 16x4 layout: K={0,1} / {2,3} halves
      v2f a0 = *(const v2f*)(Ar0 + ko);
      v2f a1 = *(const v2f*)(Ar1 + ko);
      v2f b0 = *(const v2f*)(Br0 + ko);
      v2f b1 = *(const v2f*)(Br1 + ko);
      acc00 = __builtin_amdgcn_wmma_f32_16x16x4_f32(false, a0, false, b0,
                                                    (short)0, acc00, false, false);
      acc01 = __builtin_amdgcn_wmma_f32_16x16x4_f32(false, a0, false, b1,
                                                    (short)0, acc01, false, false);
      acc10 = __builtin_amdgcn_wmma_f32_16x16x4_f32(false, a1, false, b0,
                                                    (short)0, acc10, false, false);
      acc11 = __builtin_amdgcn_wmma_f32_16x16x4_f32(false, a1, false, b1,
                                                    (short)0, acc11, false, false);
    }
  }

  // Exact diagonal dot products (only waves sitting on the diagonal need them).
  float dg0 = 0.f, dg1 = 0.f;
  if (i0 == j0) {
    const int b = bh >> 3, h = bh & 7;
    const size_t base = ((size_t)b * LSEQ + i0 + m) * DMODEL + (size_t)h * DH;
    const float* __restrict__ Cp = Ci + base;
    const float* __restrict__ Bp = Bi + base;
#pragma unroll 8
    for (int dd = 0; dd < DH; ++dd) {
      dg0 += Cp[dd] * Bp[dd];
      dg1 += Cp[dd + 16 * DMODEL] * Bp[dd + 16 * DMODEL];
    }
  }

  // Store: sub-tiles are fully-lower, fully-upper, or exactly diagonal.
  auto store_tile = [&](const v8f& acc, int i0s, int j0s, float dg) {
    const bool upper = (j0s > i0s);
    const bool diag  = (j0s == i0s);
#pragma unroll
    for (int r = 0; r < 8; ++r) {
      const int M = r + half * 8;       // C/D layout: lanes16-31 hold M+8
      float v = acc[r];
      if (upper) {
        v = 0.f;
      } else if (diag) {
        if (m > M)       v = 0.f;       // strictly upper inside diagonal tile
        else if (m == M) v = dg;        // exact C_i . B_i
      }
      outp[(size_t)(i0s + M) * LSEQ + (j0s + m)] = v;
    }
  };
  store_tile(acc00, i0,      j0,      dg0);
  store_tile(acc01, i0,      j0 + 16, dg0);
  store_tile(acc10, i0 + 16, j0,      dg1);
  store_tile(acc11, i0 + 16, j0 + 16, dg1);
}

extern "C" void kernel_launch(void* const* d_in, const int* in_sizes, int n_in,
                              void* d_out, int out_size, void* d_ws, size_t ws_size,
                              hipStream_t stream) {
  (void)in_sizes; (void)n_in; (void)out_size; (void)ws_size;
  const float* Bi = (const float*)d_in[0];   // B
  const float* Ci = (const float*)d_in[1];   // C
  const float* Si = (const float*)d_in[2];   // S (gates)
  float* Anum = (float*)d_ws;                              // 8 MB
  float* Bdiv = Anum + (size_t)NBH * LSEQ * DH;            // 8 MB
  float* Out  = (float*)d_out;

  tokenmix_prepass<<<dim3(NBH), dim3(DH), 0, stream>>>(Bi, Ci, Si, Anum, Bdiv);
  tokenmix_gemm<<<dim3(LSEQ / 64, LSEQ / 128, NBH), dim3(256), 0, stream>>>(
      Bi, Ci, Anum, Bdiv, Out);
}